// GAGNN_45054206935603
// MI455X (gfx1250) — compile-verified
//
#include <hip/hip_runtime.h>
#include <hip/hip_bf16.h>
#include <math.h>

typedef float v2f __attribute__((ext_vector_type(2)));
typedef float v8f __attribute__((ext_vector_type(8)));

__device__ __forceinline__ v8f wmma4(v2f a, v2f b, v8f c) {
  // V_WMMA_F32_16X16X4_F32 : D = A(16x4) * B(4x16) + C(16x16), fp32 exact
  return __builtin_amdgcn_wmma_f32_16x16x4_f32(false, a, false, b, (short)0, c, false, false);
}

__device__ __forceinline__ int iclamp(int v, int hi) { return v < hi ? v : hi; }

// ---------------------------------------------------------------------------
// elementwise / gather kernels
// ---------------------------------------------------------------------------

__global__ void k_gather_xenc(const float* __restrict__ feat, float* __restrict__ xenc) {
  int p = blockIdx.x * 256 + threadIdx.x;
  if (p >= 500 * 96 * 32) return;
  int e = p & 31;
  int t = p >> 5;
  int s = t / 96, n = t % 96;
  int idx = n * 500 + s;            // faithful reshape of (B,CITY,HIST,IN)->(96,500,IN)
  int b = idx / 6000, rem = idx % 6000;
  int c = rem / 12, h = rem % 12;
  xenc[p] = feat[(((b * 12 + h) * 500 + c) << 5) + e];
}

__global__ void k_wsm(const float* __restrict__ w, float* __restrict__ wsm,
                      float* __restrict__ wsmT, float* __restrict__ wrT) {
  int c = blockIdx.x * 64 + threadIdx.x;
  if (c >= 500) return;
  float v[16]; float mx = -1e30f;
  for (int g = 0; g < 16; ++g) { v[g] = w[c * 16 + g]; mx = fmaxf(mx, v[g]); }
  float s = 0.f;
  for (int g = 0; g < 16; ++g) { v[g] = __expf(v[g] - mx); s += v[g]; }
  float inv = 1.f / s;
  for (int g = 0; g < 16; ++g) {
    float sv = v[g] * inv;
    wsm[c * 16 + g] = sv;
    wsmT[g * 500 + c] = sv;
    wrT[g * 500 + c] = w[c * 16 + g];
  }
}

__global__ void k_loc(const float* __restrict__ cl, const float* __restrict__ Wl,
                      const float* __restrict__ bl, float* __restrict__ locE) {
  int p = blockIdx.x * 256 + threadIdx.x;
  if (p >= 500 * 12) return;
  int c = p / 12, j = p % 12;
  locE[p] = cl[c * 2] * Wl[j] + cl[c * 2 + 1] * Wl[12 + j] + bl[j];
}

__global__ void k_uem(const int* __restrict__ tf, const float* __restrict__ u1,
                      const float* __restrict__ u2, const float* __restrict__ u3,
                      float* __restrict__ uem) {
  int p = threadIdx.x;
  if (p >= 96) return;
  int b = p / 12, j = p % 12;
  float v;
  if (j < 4)      v = u1[(tf[b * 4 + 3] - 1) * 4 + j];        // month
  else if (j < 8) v = u2[(tf[b * 4 + 2] - 1) * 4 + (j - 4)];  // day
  else            v = u3[(tf[b * 4 + 1]) * 4 + (j - 8)];      // hour
  uem[p] = v;
}

__global__ void k_xloc(const float* __restrict__ xemb, const float* __restrict__ locE,
                       float* __restrict__ xloc) {
  int p = blockIdx.x * 256 + threadIdx.x;
  if (p >= 8 * 500 * 44) return;
  int f = p % 44; int bc = p / 44; int c = bc % 500;
  xloc[p] = (f < 32) ? xemb[bc * 32 + f] : locE[c * 12 + (f - 32)];
}

__global__ void k_gin(const float* __restrict__ gx, const float* __restrict__ uem,
                      float* __restrict__ gin) {
  int p = blockIdx.x * 256 + threadIdx.x;
  if (p >= 1920 * 100) return;
  int k = p % 100; int m = p / 100; int b = m / 240; int e = m % 240;
  int src = e / 15; int jj = e % 15; int dst = jj + (jj >= src ? 1 : 0);
  float v;
  if (k < 44)      v = gx[(b * 16 + src) * 44 + k];
  else if (k < 88) v = gx[(b * 16 + dst) * 44 + (k - 44)];
  else             v = uem[b * 12 + (k - 88)];
  gin[p] = v;
}

__global__ void k_gedges(int* __restrict__ grow, int* __restrict__ gcol) {
  int m = blockIdx.x * 256 + threadIdx.x;
  if (m >= 1920) return;
  int b = m / 240, e = m % 240;
  int src = e / 15; int jj = e % 15; int dst = jj + (jj >= src ? 1 : 0);
  grow[m] = b * 16 + src;
  gcol[m] = b * 16 + dst;
}

__global__ void k_gledges(const int* __restrict__ ei, const float* __restrict__ ewi,
                          int* __restrict__ grow, int* __restrict__ gcol,
                          float* __restrict__ ew) {
  int m = blockIdx.x * 256 + threadIdx.x;
  if (m >= 128000) return;
  int b = m / 16000, e = m % 16000;
  grow[m] = ei[e] + b * 500;
  gcol[m] = ei[16000 + e] + b * 500;
  ew[m] = ewi[e];
}

__global__ void k_cat64(const float* __restrict__ A32, const float* __restrict__ B32,
                        float* __restrict__ Y, int M) {
  int p = blockIdx.x * 256 + threadIdx.x;
  if (p >= M * 64) return;
  int f = p & 63, m = p >> 6;
  Y[p] = (f < 32) ? A32[m * 32 + f] : B32[m * 32 + (f - 32)];
}

// out[row'] = LN(A[row]+B[row]) * g + beta ; swap: (s,n) -> (n,s) row remap
__global__ void k_add_ln(const float* __restrict__ A, const float* __restrict__ Bm,
                         const float* __restrict__ g, const float* __restrict__ beta,
                         float* __restrict__ out, int swap) {
  int r = blockIdx.x;
  int e = threadIdx.x;                 // 32 lanes = 32 features
  float v = A[r * 32 + e] + Bm[r * 32 + e];
  float m = v;
  for (int off = 16; off; off >>= 1) m += __shfl_xor(m, off, 32);
  m *= (1.f / 32.f);
  float d = v - m;
  float var = d * d;
  for (int off = 16; off; off >>= 1) var += __shfl_xor(var, off, 32);
  var *= (1.f / 32.f);
  float y = d * rsqrtf(var + 1e-5f) * g[e] + beta[e];
  int ro = r;
  if (swap) { int s = r / 96, n = r % 96; ro = n * 500 + s; }
  out[ro * 32 + e] = y;
}

__global__ void k_scatter(const float* __restrict__ H, const int* __restrict__ colIdx,
                          float* __restrict__ S, int* __restrict__ cnt, int E) {
  int p = blockIdx.x * 256 + threadIdx.x;
  if (p >= E * 32) return;
  int e = p >> 5, f = p & 31;
  int c = colIdx[e];
  atomicAdd(&S[c * 32 + f], H[p]);
  if (f == 0) atomicAdd(&cnt[c], 1);
}

// ---------------------------------------------------------------------------
// fast WMMA GEMM: requires K%4==0 && N%16==0. Branch-free inner loop, v2f A loads.
// C[b] = act(A[b](MxK) * Bw[b](KxN) + bias)
// grid.x = ceil(M/16)*(N/16), grid.y = batch, block = 32 (one wave)
// ---------------------------------------------------------------------------
__global__ void k_gemm_f(const float* __restrict__ A, const float* __restrict__ Bw,
                         const float* __restrict__ bias, float* __restrict__ C,
                         int M, int N, int K, long sA, long sB, long sC, int relu) {
  int tilesN = N >> 4;
  int tm = blockIdx.x / tilesN, tn = blockIdx.x % tilesN;
  A += (long)blockIdx.y * sA;
  Bw += (long)blockIdx.y * sB;
  C += (long)blockIdx.y * sC;
  int lane = threadIdx.x, half = lane >> 4, l16 = lane & 15;
  int m0 = tm << 4, n0 = tn << 4;
  int mrow = iclamp(m0 + l16, M - 1);
  const float* Arow = A + (long)mrow * K + 2 * half;  // 8B-aligned (K even)
  const float* Bcol = Bw + n0 + l16;
  v8f acc = {};
  for (int kb = 0; kb < K; kb += 4) {
    v2f a = *(const v2f*)(Arow + kb);
    v2f b;
    b.x = Bcol[(long)(kb + 2 * half) * N];
    b.y = Bcol[(long)(kb + 2 * half + 1) * N];
    acc = wmma4(a, b, acc);
  }
  int bcol = n0 + l16;
  float bv = bias ? bias[bcol] : 0.f;
  for (int r = 0; r < 8; ++r) {
    int row = m0 + r + 8 * half;
    if (row < M) {
      float v = acc[r] + bv;
      if (relu) v = fmaxf(v, 0.f);
      C[(long)row * N + bcol] = v;
    }
  }
}

// safe WMMA GEMM: any M,N,K. Loads clamped in-bounds + cndmask select (no branches).
__global__ void k_gemm_s(const float* __restrict__ A, const float* __restrict__ Bw,
                         const float* __restrict__ bias, float* __restrict__ C,
                         int M, int N, int K, long sA, long sB, long sC, int relu) {
  int tilesN = (N + 15) >> 4;
  int tm = blockIdx.x / tilesN, tn = blockIdx.x % tilesN;
  A += (long)blockIdx.y * sA;
  Bw += (long)blockIdx.y * sB;
  C += (long)blockIdx.y * sC;
  int lane = threadIdx.x, half = lane >> 4, l16 = lane & 15;
  int m0 = tm << 4, n0 = tn << 4;
  int mrow = iclamp(m0 + l16, M - 1);
  const float* Arow = A + (long)mrow * K;
  int bcol = n0 + l16;
  int bcc = iclamp(bcol, N - 1);
  bool colok = bcol < N;
  v8f acc = {};
  for (int kb = 0; kb < K; kb += 4) {
    int ka = kb + 2 * half;
    int k0 = iclamp(ka, K - 1), k1 = iclamp(ka + 1, K - 1);
    float ax = Arow[k0];
    float ay = Arow[k1];
    float bx = Bw[(long)k0 * N + bcc];
    float by = Bw[(long)k1 * N + bcc];
    v2f a, b;
    a.x = (ka < K) ? ax : 0.f;
    a.y = (ka + 1 < K) ? ay : 0.f;
    b.x = (colok && ka < K) ? bx : 0.f;
    b.y = (colok && ka + 1 < K) ? by : 0.f;
    acc = wmma4(a, b, acc);
  }
  if (colok) {
    float bv = bias ? bias[bcol] : 0.f;
    for (int r = 0; r < 8; ++r) {
      int row = m0 + r + 8 * half;
      if (row < M) {
        float v = acc[r] + bv;
        if (relu) v = fmaxf(v, 0.f);
        C[(long)row * N + bcol] = v;
      }
    }
  }
}

// ---------------------------------------------------------------------------
// fused attention: one wave per (head nh, 16-query tile). q,k,v: [(s*96+n)*32 + h*8+d]
// ---------------------------------------------------------------------------
__global__ void k_attn(const float* __restrict__ Q, const float* __restrict__ Km,
                       const float* __restrict__ V, float* __restrict__ O) {
  __shared__ float P[16][512];
  int nh = blockIdx.y; int n = nh >> 2; int h = nh & 3;
  int m0 = blockIdx.x << 4;
  int lane = threadIdx.x, half = lane >> 4, l16 = lane & 15;
  const float scale = 0.35355339059327373f;  // 1/sqrt(8)
  int mrow = iclamp(m0 + l16, 499);
  const float* qrow = Q + ((long)mrow * 96 + n) * 32 + h * 8;
  v2f qa0 = *(const v2f*)(qrow + 2 * half);
  v2f qa1 = *(const v2f*)(qrow + 4 + 2 * half);
  for (int t0 = 0; t0 < 512; t0 += 16) {
    int tcol = t0 + l16;
    int tcc = iclamp(tcol, 499);
    const float* krow = Km + ((long)tcc * 96 + n) * 32 + h * 8;
    v2f b0 = *(const v2f*)(krow + 2 * half);       // clamped row; garbage masked below
    v2f b1 = *(const v2f*)(krow + 4 + 2 * half);
    v8f acc = {};
    acc = wmma4(qa0, b0, acc);
    acc = wmma4(qa1, b1, acc);
    for (int r = 0; r < 8; ++r)
      P[r + 8 * half][t0 + l16] = (tcol < 500) ? acc[r] * scale : -1e30f;
  }
  __syncthreads();
  if (lane < 16) {  // one lane per query row: softmax over 500 keys
    float mx = -1e30f;
    for (int t = 0; t < 500; ++t) mx = fmaxf(mx, P[lane][t]);
    float sum = 0.f;
    for (int t = 0; t < 500; ++t) { float e = __expf(P[lane][t] - mx); P[lane][t] = e; sum += e; }
    float inv = 1.f / sum;
    for (int t = 0; t < 500; ++t) P[lane][t] *= inv;
    for (int t = 500; t < 512; ++t) P[lane][t] = 0.f;
  }
  __syncthreads();
  int vc = l16 & 7;             // clamped V column; lanes 8..15 masked to zero
  bool vok = l16 < 8;
  v8f o = {};
  for (int kb = 0; kb < 500; kb += 4) {
    int ka = kb + 2 * half;
    v2f a = *(const v2f*)(&P[l16][ka]);
    float bx = V[((long)ka * 96 + n) * 32 + h * 8 + vc];
    float by = V[((long)(ka + 1) * 96 + n) * 32 + h * 8 + vc];
    v2f b;
    b.x = vok ? bx : 0.f;
    b.y = vok ? by : 0.f;
    o = wmma4(a, b, o);
  }
  if (vok) {
    for (int r = 0; r < 8; ++r) {
      int row = m0 + r + 8 * half;
      if (row < 500) O[((long)row * 96 + n) * 32 + h * 8 + l16] = o[r];
    }
  }
}

// ---------------------------------------------------------------------------
// fused feed-forward: F = relu(X*W1+b1)*W2+b2  (16 rows per wave, mid in LDS)
// ---------------------------------------------------------------------------
__global__ void k_ff(const float* __restrict__ X, const float* __restrict__ W1,
                     const float* __restrict__ b1, const float* __restrict__ W2,
                     const float* __restrict__ b2, float* __restrict__ F, int M) {
  __shared__ float MID[16][256];
  int lane = threadIdx.x, half = lane >> 4, l16 = lane & 15;
  int m0 = blockIdx.x << 4;
  int mr = iclamp(m0 + l16, M - 1);
  const float* xr = X + (long)mr * 32 + 2 * half;
  for (int nt = 0; nt < 16; ++nt) {
    int col = nt * 16 + l16;
    v8f acc = {};
    for (int kb = 0; kb < 32; kb += 4) {
      int ka = kb + 2 * half;
      v2f a = *(const v2f*)(xr + kb);
      v2f b; b.x = W1[ka * 256 + col]; b.y = W1[(ka + 1) * 256 + col];
      acc = wmma4(a, b, acc);
    }
    float bb = b1[col];
    for (int r = 0; r < 8; ++r) MID[r + 8 * half][col] = fmaxf(acc[r] + bb, 0.f);
  }
  __syncthreads();
  for (int nt = 0; nt < 2; ++nt) {
    int col = nt * 16 + l16;
    v8f acc = {};
    for (int kb = 0; kb < 256; kb += 4) {
      int ka = kb + 2 * half;
      v2f a = *(const v2f*)(&MID[l16][ka]);
      v2f b; b.x = W2[ka * 32 + col]; b.y = W2[(ka + 1) * 32 + col];
      acc = wmma4(a, b, acc);
    }
    float bb = b2[col];
    for (int r = 0; r < 8; ++r) {
      int row = m0 + r + 8 * half;
      if (row < M) F[(long)row * 32 + col] = acc[r] + bb;
    }
  }
}

// ---------------------------------------------------------------------------
// edge MLP1: H[e] = relu([X[rowIdx[e]] (nh) | EA[e] (wa)] * W + b), N=32
// nh % 4 == 0 required; wa must be 12 or 1.
// ---------------------------------------------------------------------------
__global__ void k_edge_mlp1(const float* __restrict__ X, int ldx, int nh,
                            const int* __restrict__ rowIdx,
                            const float* __restrict__ EA, int wa,
                            const float* __restrict__ Wt, const float* __restrict__ bias,
                            float* __restrict__ H, int E) {
  int tn = blockIdx.x & 1, tm = blockIdx.x >> 1;
  int lane = threadIdx.x, half = lane >> 4, l16 = lane & 15;
  int m0 = tm << 4, n0 = tn << 4;
  int m = iclamp(m0 + l16, E - 1);
  int r0 = rowIdx[m];
  int col = n0 + l16;
  const float* Xr = X + (long)r0 * ldx + 2 * half;
  const float* Wc = Wt + col;
  v8f acc = {};
  for (int kb = 0; kb < nh; kb += 4) {
    int ka = kb + 2 * half;
    v2f a = *(const v2f*)(Xr + kb);
    v2f b; b.x = Wc[ka * 32]; b.y = Wc[(ka + 1) * 32];
    acc = wmma4(a, b, acc);
  }
  if (wa == 12) {
    const float* Er = EA + (long)m * 12 + 2 * half;
    for (int kb = 0; kb < 12; kb += 4) {
      int ka = nh + kb + 2 * half;
      v2f a = *(const v2f*)(Er + kb);
      v2f b; b.x = Wc[ka * 32]; b.y = Wc[(ka + 1) * 32];
      acc = wmma4(a, b, acc);
    }
  } else {  // wa == 1: single valid K row (nh) lives in half 0, slot x
    float ea = EA[m];
    float wv = Wc[nh * 32];
    v2f a, b;
    a.x = (half == 0) ? ea : 0.f;  a.y = 0.f;
    b.x = (half == 0) ? wv : 0.f;  b.y = 0.f;
    acc = wmma4(a, b, acc);
  }
  float bb = bias[col];
  for (int r = 0; r < 8; ++r) {
    int row = m0 + r + 8 * half;
    if (row < E) H[(long)row * 32 + col] = fmaxf(acc[r] + bb, 0.f);
  }
}

// ---------------------------------------------------------------------------
// node MLP2: Y[m] = relu([X[m] (nh) | S[m]/max(cnt[m],1) (32)] * W + b), N=32
// nh % 4 == 0 required.
// ---------------------------------------------------------------------------
__global__ void k_node_mlp2(const float* __restrict__ X, int nh,
                            const float* __restrict__ S, const int* __restrict__ cnt,
                            const float* __restrict__ Wt, const float* __restrict__ bias,
                            float* __restrict__ Y, int M) {
  int tn = blockIdx.x & 1, tm = blockIdx.x >> 1;
  int lane = threadIdx.x, half = lane >> 4, l16 = lane & 15;
  int m0 = tm << 4, n0 = tn << 4;
  int m = iclamp(m0 + l16, M - 1);
  float ic = 1.f / fmaxf((float)cnt[m], 1.f);
  int col = n0 + l16;
  const float* Xr = X + (long)m * nh + 2 * half;
  const float* Sr = S + (long)m * 32 + 2 * half;
  const float* Wc = Wt + col;
  v8f acc = {};
  for (int kb = 0; kb < nh; kb += 4) {
    int ka = kb + 2 * half;
    v2f a = *(const v2f*)(Xr + kb);
    v2f b; b.x = Wc[ka * 32]; b.y = Wc[(ka + 1) * 32];
    acc = wmma4(a, b, acc);
  }
  for (int kb = 0; kb < 32; kb += 4) {
    int ka = nh + kb + 2 * half;
    v2f a = *(const v2f*)(Sr + kb);
    a.x *= ic; a.y *= ic;
    v2f b; b.x = Wc[ka * 32]; b.y = Wc[(ka + 1) * 32];
    acc = wmma4(a, b, acc);
  }
  float bb = bias[col];
  for (int r = 0; r < 8; ++r) {
    int row = m0 + r + 8 * half;
    if (row < M) Y[(long)row * 32 + col] = fmaxf(acc[r] + bb, 0.f);
  }
}

// ---------------------------------------------------------------------------
// host orchestration
// ---------------------------------------------------------------------------
enum {
  P_DEC_GGNN0_M1_W = 6, P_DEC_GGNN0_M1_B, P_DEC_GGNN0_M2_W, P_DEC_GGNN0_M2_B,
  P_DEC_GGNN1_M1_W, P_DEC_GGNN1_M1_B, P_DEC_GGNN1_M2_W, P_DEC_GGNN1_M2_B,
  P_DEC_GRP0_M1_W, P_DEC_GRP0_M1_B, P_DEC_GRP0_M2_W, P_DEC_GRP0_M2_B,
  P_DEC_GRP1_M1_W, P_DEC_GRP1_M1_B, P_DEC_GRP1_M2_W, P_DEC_GRP1_M2_B,
  P_DEC_XEMB_W, P_DEC_XEMB_B,
  P_EDGEINF_W, P_EDGEINF_B,
  P_ENC_WK, P_ENC_WO, P_ENC_WQ, P_ENC_WV,
  P_ENC_BK, P_ENC_BO, P_ENC_BQ, P_ENC_BV,
  P_ENC_FF1_W, P_ENC_FF1_B, P_ENC_FF2_W, P_ENC_FF2_B,
  P_ENC_LN1_B, P_ENC_LN1_G, P_ENC_LN2_B, P_ENC_LN2_G,
  P_GLB0_M1_W, P_GLB0_M1_B, P_GLB0_M2_W, P_GLB0_M2_B,
  P_GLB1_M1_W, P_GLB1_M1_B, P_GLB1_M2_W, P_GLB1_M2_B,
  P_GRP0_M1_W, P_GRP0_M1_B, P_GRP0_M2_W, P_GRP0_M2_B,
  P_GRP1_M1_W, P_GRP1_M1_B, P_GRP1_M2_W, P_GRP1_M2_B,
  P_LOC_W, P_LOC_B,
  P_PRED1_W, P_PRED1_B, P_PRED2_W, P_PRED2_B,
  P_U1, P_U2, P_U3, P_W, P_XEMB_W, P_XEMB_B, P_COUNT
};

extern "C" void kernel_launch(void* const* d_in, const int* in_sizes, int n_in,
                              void* d_out, int out_size, void* d_ws, size_t ws_size,
                              hipStream_t stream) {
  (void)in_sizes; (void)out_size; (void)ws_size;
  const float* feature  = (const float*)d_in[1];
  const int*   tf       = (const int*)d_in[2];
  const int*   eidx     = (const int*)d_in[3];
  const float* ewi      = (const float*)d_in[4];
  const float* city_loc = (const float*)d_in[5];
  const float* P[P_COUNT];
  for (int i = 0; i < P_COUNT && i < n_in; ++i) P[i] = (const float*)d_in[i];

  float* W = (float*)d_ws;
  // ---- encoder-region arena (dead after x_emb) ----
  const size_t o_xenc = 0, o_q = 1536000, o_k = 3072000, o_v = 4608000,
               o_obuf = 6144000, o_proj = 7680000, o_x1 = 9216000,
               o_ffo = 10752000, o_x2 = 12288000;
  // ---- aliased GNN buffers (live only after encoder) ----
  const size_t o_bigH = 0, o_n64 = 4096000, o_GS = 4352000, o_gA = 4480000,
               o_gB = 4608000, o_dx = 4736000, o_p1 = 4864000;
  // ---- persistent region ----
  const size_t o_xemb = 13824000, o_wsm = 13952000, o_wsmT = 13960000,
               o_wrT = 13968000, o_locE = 13976000, o_uem = 13982000,
               o_xloc = 13982128, o_gx = 14158128, o_gin = 14163760,
               o_gewf = 14355760, o_grow = 14378800, o_gcol = 14380720,
               o_gH = 14382640, o_gS = 14444080, o_gcnt = 14448176,
               o_gnA = 14448304, o_gnB = 14452400, o_newx = 14456496,
               o_gerow = 14584496, o_gecol = 14712496, o_ewf = 14840496,
               o_Gcnt = 14968496;

  int* grow  = (int*)(W + o_grow);
  int* gcol  = (int*)(W + o_gcol);
  int* gcnt  = (int*)(W + o_gcnt);
  int* gerow = (int*)(W + o_gerow);
  int* gecol = (int*)(W + o_gecol);
  int* Gcnt  = (int*)(W + o_Gcnt);

  // ===== encoder =====
  k_gather_xenc<<<6000, 256, 0, stream>>>(feature, W + o_xenc);
  k_gemm_f<<<dim3(6000, 1), 32, 0, stream>>>(W + o_xenc, P[P_ENC_WQ], P[P_ENC_BQ], W + o_q, 48000, 32, 32, 0, 0, 0, 0);
  k_gemm_f<<<dim3(6000, 1), 32, 0, stream>>>(W + o_xenc, P[P_ENC_WK], P[P_ENC_BK], W + o_k, 48000, 32, 32, 0, 0, 0, 0);
  k_gemm_f<<<dim3(6000, 1), 32, 0, stream>>>(W + o_xenc, P[P_ENC_WV], P[P_ENC_BV], W + o_v, 48000, 32, 32, 0, 0, 0, 0);
  k_attn<<<dim3(32, 384), 32, 0, stream>>>(W + o_q, W + o_k, W + o_v, W + o_obuf);
  k_gemm_f<<<dim3(6000, 1), 32, 0, stream>>>(W + o_obuf, P[P_ENC_WO], P[P_ENC_BO], W + o_proj, 48000, 32, 32, 0, 0, 0, 0);
  k_add_ln<<<48000, 32, 0, stream>>>(W + o_xenc, W + o_proj, P[P_ENC_LN1_G], P[P_ENC_LN1_B], W + o_x1, 1);
  k_ff<<<3000, 32, 0, stream>>>(W + o_x1, P[P_ENC_FF1_W], P[P_ENC_FF1_B], P[P_ENC_FF2_W], P[P_ENC_FF2_B], W + o_ffo, 48000);
  k_add_ln<<<48000, 32, 0, stream>>>(W + o_x1, W + o_ffo, P[P_ENC_LN2_G], P[P_ENC_LN2_B], W + o_x2, 0);
  // x_embed: (4000,384)@(384,32)
  k_gemm_f<<<dim3(500, 1), 32, 0, stream>>>(W + o_x2, P[P_XEMB_W], P[P_XEMB_B], W + o_xemb, 4000, 32, 384, 0, 0, 0, 0);

  // ===== group graph (encoder side) =====
  k_wsm<<<8, 64, 0, stream>>>(P[P_W], W + o_wsm, W + o_wsmT, W + o_wrT);
  k_loc<<<24, 256, 0, stream>>>(city_loc, P[P_LOC_W], P[P_LOC_B], W + o_locE);
  k_uem<<<1, 128, 0, stream>>>(tf, P[P_U1], P[P_U2], P[P_U3], W + o_uem);
  k_xloc<<<688, 256, 0, stream>>>(W + o_xemb, W + o_locE, W + o_xloc);
  // g_x[b] = wsmT(16x500) @ xloc[b](500x44)   (N=44 -> safe path)
  k_gemm_s<<<dim3(3, 8), 32, 0, stream>>>(W + o_wsmT, W + o_xloc, nullptr, W + o_gx, 16, 44, 500, 0, 22000, 704, 0);
  k_gedges<<<8, 256, 0, stream>>>(grow, gcol);
  k_gin<<<750, 256, 0, stream>>>(W + o_gx, W + o_uem, W + o_gin);
  k_gemm_s<<<dim3(120, 1), 32, 0, stream>>>(W + o_gin, P[P_EDGEINF_W], P[P_EDGEINF_B], W + o_gewf, 1920, 12, 100, 0, 0, 0, 1);
  // group gnn layer 0 (node_h=44)
  k_edge_mlp1<<<240, 32, 0, stream>>>(W + o_gx, 44, 44, grow, W + o_gewf, 12, P[P_GRP0_M1_W], P[P_GRP0_M1_B], W + o_gH, 1920);
  hipMemsetAsync(W + o_gS, 0, 4096 * 4, stream); hipMemsetAsync(gcnt, 0, 128 * 4, stream);
  k_scatter<<<240, 256, 0, stream>>>(W + o_gH, gcol, W + o_gS, gcnt, 1920);
  k_node_mlp2<<<16, 32, 0, stream>>>(W + o_gx, 44, W + o_gS, gcnt, P[P_GRP0_M2_W], P[P_GRP0_M2_B], W + o_gnA, 128);
  // group gnn layer 1 (node_h=32)
  k_edge_mlp1<<<240, 32, 0, stream>>>(W + o_gnA, 32, 32, grow, W + o_gewf, 12, P[P_GRP1_M1_W], P[P_GRP1_M1_B], W + o_gH, 1920);
  hipMemsetAsync(W + o_gS, 0, 4096 * 4, stream); hipMemsetAsync(gcnt, 0, 128 * 4, stream);
  k_scatter<<<240, 256, 0, stream>>>(W + o_gH, gcol, W + o_gS, gcnt, 1920);
  k_node_mlp2<<<16, 32, 0, stream>>>(W + o_gnA, 32, W + o_gS, gcnt, P[P_GRP1_M2_W], P[P_GRP1_M2_B], W + o_gnB, 128);
  // new_x[b] = wsm(500x16) @ g_xf[b](16x32)   (K=16, N=32 -> fast; M=500 handled by clamp)
  k_gemm_f<<<dim3(64, 8), 32, 0, stream>>>(W + o_wsm, W + o_gnB, nullptr, W + o_newx, 500, 32, 16, 0, 512, 16000, 0);
  k_cat64<<<1000, 256, 0, stream>>>(W + o_xemb, W + o_newx, W + o_n64, 4000);

  // ===== global graph (encoder side) =====
  k_gledges<<<500, 256, 0, stream>>>(eidx, ewi, gerow, gecol, W + o_ewf);
  // layer 0 (node_h=64, K=65)
  k_edge_mlp1<<<16000, 32, 0, stream>>>(W + o_n64, 64, 64, gerow, W + o_ewf, 1, P[P_GLB0_M1_W], P[P_GLB0_M1_B], W + o_bigH, 128000);
  hipMemsetAsync(W + o_GS, 0, 128000 * 4, stream); hipMemsetAsync(Gcnt, 0, 4000 * 4, stream);
  k_scatter<<<16000, 256, 0, stream>>>(W + o_bigH, gecol, W + o_GS, Gcnt, 128000);
  k_node_mlp2<<<500, 32, 0, stream>>>(W + o_n64, 64, W + o_GS, Gcnt, P[P_GLB0_M2_W], P[P_GLB0_M2_B], W + o_gA, 4000);
  // layer 1 (node_h=32, K=33)
  k_edge_mlp1<<<16000, 32, 0, stream>>>(W + o_gA, 32, 32, gerow, W + o_ewf, 1, P[P_GLB1_M1_W], P[P_GLB1_M1_B], W + o_bigH, 128000);
  hipMemsetAsync(W + o_GS, 0, 128000 * 4, stream); hipMemsetAsync(Gcnt, 0, 4000 * 4, stream);
  k_scatter<<<16000, 256, 0, stream>>>(W + o_bigH, gecol, W + o_GS, Gcnt, 128000);
  k_node_mlp2<<<500, 32, 0, stream>>>(W + o_gA, 32, W + o_GS, Gcnt, P[P_GLB1_M2_W], P[P_GLB1_M2_B], W + o_gB, 4000);

  // ===== decoder =====
  k_gemm_f<<<dim3(500, 1), 32, 0, stream>>>(W + o_gB, P[P_DEC_XEMB_W], P[P_DEC_XEMB_B], W + o_dx, 4000, 32, 32, 0, 0, 0, 0);
  // dg[b] = wrT(16x500) @ dx[b](500x32)
  k_gemm_f<<<dim3(2, 8), 32, 0, stream>>>(W + o_wrT, W + o_dx, nullptr, W + o_gnA, 16, 32, 500, 0, 16000, 512, 0);
  // dec group gnn layer 0 (node_h=32)
  k_edge_mlp1<<<240, 32, 0, stream>>>(W + o_gnA, 32, 32, grow, W + o_gewf, 12, P[P_DEC_GRP0_M1_W], P[P_DEC_GRP0_M1_B], W + o_gH, 1920);
  hipMemsetAsync(W + o_gS, 0, 4096 * 4, stream); hipMemsetAsync(gcnt, 0, 128 * 4, stream);
  k_scatter<<<240, 256, 0, stream>>>(W + o_gH, gcol, W + o_gS, gcnt, 1920);
  k_node_mlp2<<<16, 32, 0, stream>>>(W + o_gnA, 32, W + o_gS, gcnt, P[P_DEC_GRP0_M2_W], P[P_DEC_GRP0_M2_B], W + o_gnB, 128);
  // dec group gnn layer 1
  k_edge_mlp1<<<240, 32, 0, stream>>>(W + o_gnB, 32, 32, grow, W + o_gewf, 12, P[P_DEC_GRP1_M1_W], P[P_DEC_GRP1_M1_B], W + o_gH, 1920);
  hipMemsetAsync(W + o_gS, 0, 4096 * 4, stream); hipMemsetAsync(gcnt, 0, 128 * 4, stream);
  k_scatter<<<240, 256, 0, stream>>>(W + o_gH, gcol, W + o_gS, gcnt, 1920);
  k_node_mlp2<<<16, 32, 0, stream>>>(W + o_gnB, 32, W + o_gS, gcnt, P[P_DEC_GRP1_M2_W], P[P_DEC_GRP1_M2_B], W + o_gnA, 128);
  // dn[b] = w_raw(500x16) @ dg[b](16x32)
  k_gemm_f<<<dim3(64, 8), 32, 0, stream>>>(P[P_W], W + o_gnA, nullptr, W + o_newx, 500, 32, 16, 0, 512, 16000, 0);
  k_cat64<<<1000, 256, 0, stream>>>(W + o_dx, W + o_newx, W + o_n64, 4000);
  // dec global gnn layer 0
  k_edge_mlp1<<<16000, 32, 0, stream>>>(W + o_n64, 64, 64, gerow, W + o_ewf, 1, P[P_DEC_GGNN0_M1_W], P[P_DEC_GGNN0_M1_B], W + o_bigH, 128000);
  hipMemsetAsync(W + o_GS, 0, 128000 * 4, stream); hipMemsetAsync(Gcnt, 0, 4000 * 4, stream);
  k_scatter<<<16000, 256, 0, stream>>>(W + o_bigH, gecol, W + o_GS, Gcnt, 128000);
  k_node_mlp2<<<500, 32, 0, stream>>>(W + o_n64, 64, W + o_GS, Gcnt, P[P_DEC_GGNN0_M2_W], P[P_DEC_GGNN0_M2_B], W + o_gA, 4000);
  // dec global gnn layer 1
  k_edge_mlp1<<<16000, 32, 0, stream>>>(W + o_gA, 32, 32, gerow, W + o_ewf, 1, P[P_DEC_GGNN1_M1_W], P[P_DEC_GGNN1_M1_B], W + o_bigH, 128000);
  hipMemsetAsync(W + o_GS, 0, 128000 * 4, stream); hipMemsetAsync(Gcnt, 0, 4000 * 4, stream);
  k_scatter<<<16000, 256, 0, stream>>>(W + o_bigH, gecol, W + o_GS, Gcnt, 128000);
  k_node_mlp2<<<500, 32, 0, stream>>>(W + o_gA, 32, W + o_GS, Gcnt, P[P_DEC_GGNN1_M2_W], P[P_DEC_GGNN1_M2_B], W + o_gB, 4000);

  // ===== prediction head =====
  k_gemm_f<<<dim3(250, 1), 32, 0, stream>>>(W + o_gB, P[P_PRED1_W], P[P_PRED1_B], W + o_p1, 4000, 16, 32, 0, 0, 0, 1);
  k_gemm_s<<<dim3(250, 1), 32, 0, stream>>>(W + o_p1, P[P_PRED2_W], P[P_PRED2_B], (float*)d_out, 4000, 12, 16, 0, 0, 0, 1);
}